// GOT_79164837200181
// MI455X (gfx1250) — compile-verified
//
#include <hip/hip_runtime.h>
#include <math.h>

typedef __attribute__((ext_vector_type(2))) float v2f;
typedef __attribute__((ext_vector_type(4))) float v4f;
typedef __attribute__((ext_vector_type(8))) float v8f;
typedef __attribute__((ext_vector_type(4))) int   v4i;

#define TPB 256

// dims of the problem (from setup_inputs)
#define BSZ 32
#define NI 196
#define NT 256
#define DD 1024

// CDNA5 async global->LDS copy (bypasses VGPRs, tracked by ASYNCcnt)
#if defined(__has_builtin)
#if __has_builtin(__builtin_amdgcn_global_load_async_to_lds_b128)
#define HAVE_ASYNC_LDS 1
#endif
#endif
#ifndef HAVE_ASYNC_LDS
#define HAVE_ASYNC_LDS 0
#endif

#if HAVE_ASYNC_LDS
typedef __attribute__((address_space(1))) v4i* gptr_v4i;
typedef __attribute__((address_space(3))) v4i* lptr_v4i;
#endif

__device__ __forceinline__ void wait_async0() {
#if defined(__has_builtin) && __has_builtin(__builtin_amdgcn_s_wait_asynccnt)
  __builtin_amdgcn_s_wait_asynccnt(0);
#elif defined(__AMDGCN__)
  asm volatile("s_wait_asynccnt 0" ::: "memory");
#endif
}

// ---------------- reduction helper ----------------
__device__ __forceinline__ float blockReduceSum(float v, float* red) {
  int tid = threadIdx.x;
  red[tid] = v;
  __syncthreads();
  for (int s = TPB / 2; s > 0; s >>= 1) {
    if (tid < s) red[tid] += red[tid + s];
    __syncthreads();
  }
  return red[0];
}

// ---------------- row L2-normalize (optionally masked) ----------------
// grid = b*rows blocks, 256 threads; in/out: (b, rows, d) row-major
__global__ void k_norm_rows(const float* __restrict__ in, const float* __restrict__ mask,
                            float* __restrict__ out, int d) {
  __shared__ float red[TPB];
  size_t row = blockIdx.x;
  const float* src = in + row * (size_t)d;
  float mval = mask ? mask[row] : 1.0f;
  float ss = 0.f;
  for (int k = threadIdx.x; k < d; k += TPB) {
    float v = src[k] * mval;
    ss += v * v;
  }
  ss = blockReduceSum(ss, red);
  float scale = mval / (sqrtf(ss) + 1e-12f);
  float* dst = out + row * (size_t)d;
  for (int k = threadIdx.x; k < d; k += TPB) dst[k] = src[k] * scale;
}

// ---------------- WMMA f32 GEMM, D = base + alpha*(A·B^T) + rowBias + colBias ----------
// A: (b, rowsA, kdim), B: (b, rowsB, kdim), D: (b, rowsA, rowsB)
// block = 256 threads = 8 waves; block computes a 16 x 128 output strip.
// A panel (16 x 64) is staged into LDS, double-buffered with async global->LDS
// copies so the next panel's HBM fetch overlaps this panel's WMMAs.
// grid = (ceil(rowsA/16), ceil(rowsB/128), b)
__global__ void k_wmma_gemm_nt(const float* __restrict__ A, const float* __restrict__ B,
                               float* __restrict__ D, int rowsA, int rowsB, int kdim,
                               const float* __restrict__ rowBias,
                               const float* __restrict__ colBias,
                               float alpha, float base) {
  int b = blockIdx.z;
  int tileM = blockIdx.x;
  int tid = threadIdx.x, wave = tid >> 5, lane = tid & 31;
  int tileN = blockIdx.y * 8 + wave;
  int lg = lane >> 4;   // half-wave group (0/1)
  int li = lane & 15;

  const float* Ab = A + (size_t)b * rowsA * kdim;
  const float* Bb = B + (size_t)b * rowsB * kdim;

  __shared__ float As[2][16][68];  // double-buffered 16x64 panel, padded rows

  // this thread's 16B slice of the A panel
  int f = tid * 4;
  int r = f >> 6, c = f & 63;
  int arow = tileM * 16 + r;
  if (arow > rowsA - 1) arow = rowsA - 1;
  const float* Asrc = Ab + (size_t)arow * kdim + c;

  int brow = tileN * 16 + li;
  if (brow > rowsB - 1) brow = rowsB - 1;
  const float* Brow = Bb + (size_t)brow * kdim;

#if HAVE_ASYNC_LDS
  __builtin_amdgcn_global_load_async_to_lds_b128(
      (gptr_v4i)(Asrc), (lptr_v4i)(&As[0][r][c]), 0, 0);
#endif

  v8f acc = {};
  int ib = 0;
  for (int kb = 0; kb < kdim; kb += 64, ib ^= 1) {
#if HAVE_ASYNC_LDS
    wait_async0();          // this wave's slice of panel `ib` has landed in LDS
    __syncthreads();        // ... and everyone else's too
    if (kb + 64 < kdim)     // kick off next panel into the other buffer
      __builtin_amdgcn_global_load_async_to_lds_b128(
          (gptr_v4i)(Asrc + kb + 64), (lptr_v4i)(&As[ib ^ 1][r][c]), 0, 0);
#else
    *(v4f*)&As[ib][r][c] = *(const v4f*)(Asrc + kb);
    __syncthreads();
#endif
    if (kb + 64 < kdim) __builtin_prefetch(Brow + kb + 64, 0, 1);
#pragma unroll
    for (int kk = 0; kk < 64; kk += 4) {
      v2f av = *(const v2f*)&As[ib][li][kk + 2 * lg];
      v2f bv = *(const v2f*)(Brow + kb + kk + 2 * lg);
      acc = __builtin_amdgcn_wmma_f32_16x16x4_f32(false, av, false, bv,
                                                  (short)0, acc, false, false);
    }
    __syncthreads();        // all waves done reading buffer `ib` before reuse
  }

  int ncol = tileN * 16 + li;
#pragma unroll
  for (int v = 0; v < 8; v++) {
    int mrow = tileM * 16 + v + 8 * lg;
    if (mrow < rowsA && ncol < rowsB) {
      float val = base + alpha * acc[v];
      if (rowBias) val += rowBias[(size_t)b * rowsA + mrow];
      if (colBias) val += colBias[(size_t)b * rowsB + ncol];
      D[((size_t)b * rowsA + mrow) * rowsB + ncol] = val;
    }
  }
}

// ---------------- WMMA f32 GEMM (no transpose), D = A·B ----------------
// A: (b, rowsA, kdim), B: (b, kdim, colsB), D: (b, rowsA, colsB)
// block = 256 threads = 8 waves, each wave one 16x16 tile.
__global__ void k_wmma_gemm_nn(const float* __restrict__ A, const float* __restrict__ B,
                               float* __restrict__ D, int rowsA, int kdim, int colsB) {
  int b = blockIdx.z;
  int tileM = blockIdx.x;
  int tid = threadIdx.x, wave = tid >> 5, lane = tid & 31;
  int tileN = blockIdx.y * 8 + wave;
  int lg = lane >> 4, li = lane & 15;

  const float* Ab = A + (size_t)b * rowsA * kdim;
  const float* Bb = B + (size_t)b * kdim * colsB;

  int arow = tileM * 16 + li;
  if (arow > rowsA - 1) arow = rowsA - 1;
  int ncol = tileN * 16 + li;
  if (ncol > colsB - 1) ncol = colsB - 1;
  const float* Arow = Ab + (size_t)arow * kdim;

  v8f acc = {};
  for (int k = 0; k < kdim; k += 4) {
    int kk = k + 2 * lg;
    v2f av = *(const v2f*)(Arow + kk);
    v2f bv;
    bv[0] = Bb[(size_t)kk * colsB + ncol];
    bv[1] = Bb[(size_t)(kk + 1) * colsB + ncol];
    acc = __builtin_amdgcn_wmma_f32_16x16x4_f32(false, av, false, bv,
                                                (short)0, acc, false, false);
  }

  int nc = tileN * 16 + li;
#pragma unroll
  for (int v = 0; v < 8; v++) {
    int mrow = tileM * 16 + v + 8 * lg;
    if (mrow < rowsA && nc < colsB)
      D[((size_t)b * rowsA + mrow) * colsB + nc] = acc[v];
  }
}

// ---------------- global min/max -> threshold ----------------
__global__ void k_minmax_partial(const float* __restrict__ X, size_t N,
                                 float* __restrict__ pmin, float* __restrict__ pmax) {
  __shared__ float smin[TPB], smax[TPB];
  float mn = 3.4e38f, mx = -3.4e38f;
  for (size_t i = (size_t)blockIdx.x * TPB + threadIdx.x; i < N;
       i += (size_t)gridDim.x * TPB) {
    float v = X[i];
    mn = fminf(mn, v);
    mx = fmaxf(mx, v);
  }
  int tid = threadIdx.x;
  smin[tid] = mn; smax[tid] = mx;
  __syncthreads();
  for (int s = TPB / 2; s > 0; s >>= 1) {
    if (tid < s) {
      smin[tid] = fminf(smin[tid], smin[tid + s]);
      smax[tid] = fmaxf(smax[tid], smax[tid + s]);
    }
    __syncthreads();
  }
  if (tid == 0) { pmin[blockIdx.x] = smin[0]; pmax[blockIdx.x] = smax[0]; }
}

__global__ void k_minmax_final(const float* __restrict__ pmin, const float* __restrict__ pmax,
                               int cnt, float* __restrict__ thr) {
  __shared__ float smin[TPB], smax[TPB];
  float mn = 3.4e38f, mx = -3.4e38f;
  for (int i = threadIdx.x; i < cnt; i += TPB) {
    mn = fminf(mn, pmin[i]);
    mx = fmaxf(mx, pmax[i]);
  }
  int tid = threadIdx.x;
  smin[tid] = mn; smax[tid] = mx;
  __syncthreads();
  for (int s = TPB / 2; s > 0; s >>= 1) {
    if (tid < s) {
      smin[tid] = fminf(smin[tid], smin[tid + s]);
      smax[tid] = fmaxf(smax[tid], smax[tid + s]);
    }
    __syncthreads();
  }
  if (tid == 0) thr[0] = smin[0] + 0.1f * (smax[0] - smin[0]);
}

// C = relu(C - thr) [* mask?1:1e-5], in place
__global__ void k_thr_relu_mask(float* __restrict__ C, const int* __restrict__ mask,
                                const float* __restrict__ thr, size_t N) {
  float t = thr[0];
  for (size_t i = (size_t)blockIdx.x * TPB + threadIdx.x; i < N;
       i += (size_t)gridDim.x * TPB) {
    float v = C[i] - t;
    v = v > 0.f ? v : 0.f;
    if (mask) v *= (mask[i] == 1 ? 1.0f : 1e-5f);
    C[i] = v;
  }
}

// ---------------- misc elementwise ----------------
__global__ void k_exp_neg2(const float* __restrict__ C, float* __restrict__ A, size_t N) {
  for (size_t i = (size_t)blockIdx.x * TPB + threadIdx.x; i < N;
       i += (size_t)gridDim.x * TPB)
    A[i] = expf(-2.0f * C[i]);  // exp(-C / 0.5)
}

__global__ void k_fill(float* __restrict__ p, float v, size_t N) {
  for (size_t i = (size_t)blockIdx.x * TPB + threadIdx.x; i < N;
       i += (size_t)gridDim.x * TPB)
    p[i] = v;
}

// out[row] = scale * sum_k X[row,k]^2
__global__ void k_rowsumsq(const float* __restrict__ X, float* __restrict__ out,
                           int cols, float scale) {
  __shared__ float red[TPB];
  size_t row = blockIdx.x;
  const float* x = X + row * (size_t)cols;
  float s = 0.f;
  for (int k = threadIdx.x; k < cols; k += TPB) { float v = x[k]; s += v * v; }
  s = blockReduceSum(s, red);
  if (threadIdx.x == 0) out[row] = s * scale;
}

// ---------------- IPOT recursion ----------------
// Q_{k+1} = A .* delta_k .* Q_k .* sigma_k^T  (first iter: Q_1 = A)
// delta_{k+1}[i] = 1 / (n * sum_j Q_{k+1}[i,j] * sigma_k[j])
// grid = b*n blocks, 256 threads (m == 256)
__global__ void k_ipot_qdelta(const float* __restrict__ A, float* __restrict__ Q,
                              float* __restrict__ delta, const float* __restrict__ sigma,
                              int n, int m, int first) {
  __shared__ float red[TPB];
  size_t row = blockIdx.x;       // b*n + i
  size_t b = row / n;
  int j = threadIdx.x;
  size_t off = row * (size_t)m + j;
  float sg = sigma[b * (size_t)m + j];
  float a = A[off];
  float qn = first ? a : (a * delta[row] * Q[off] * sg);
  Q[off] = qn;
  float s = blockReduceSum(qn * sg, red);
  if (j == 0) delta[row] = 1.0f / ((float)n * s);
}

// sigma_k[j] = 1 / (m * sum_i Q[i,j] * delta[i]); one block per batch, coalesced rows
__global__ void k_ipot_sigma(const float* __restrict__ Q, const float* __restrict__ delta,
                             float* __restrict__ sigma, int n, int m) {
  __shared__ float dsh[256];
  size_t b = blockIdx.x;
  int j = threadIdx.x;
  for (int i = threadIdx.x; i < n; i += TPB) dsh[i] = delta[b * (size_t)n + i];
  __syncthreads();
  const float* Qb = Q + b * (size_t)n * m;
  float s = 0.f;
  for (int i = 0; i < n; i++) s += Qb[(size_t)i * m + j] * dsh[i];
  sigma[b * (size_t)m + j] = 1.0f / ((float)m * s);
}

// T = delta .* Q .* sigma^T ; grid = b*n blocks
__global__ void k_ipot_finalize(const float* __restrict__ Q, const float* __restrict__ delta,
                                const float* __restrict__ sigma, float* __restrict__ T,
                                int n, int m) {
  size_t row = blockIdx.x;
  size_t b = row / n;
  int j = threadIdx.x;
  size_t off = row * (size_t)m + j;
  T[off] = delta[row] * Q[off] * sigma[b * (size_t)m + j];
}

// out[b] = sum(X[b] .* Y[b]); one block per batch
__global__ void k_batch_dot(const float* __restrict__ X, const float* __restrict__ Y,
                            float* __restrict__ out, size_t per) {
  __shared__ float red[TPB];
  size_t b = blockIdx.x;
  const float* x = X + b * per;
  const float* y = Y + b * per;
  float s = 0.f;
  for (size_t i = threadIdx.x; i < per; i += TPB) s += x[i] * y[i];
  s = blockReduceSum(s, red);
  if (threadIdx.x == 0) out[b] = s;
}

// twd = 0.1*mean(gwd) + 0.1*mean(-wd_sum)
__global__ void k_final(const float* __restrict__ wds, const float* __restrict__ gws,
                        float* __restrict__ out0, int bs) {
  if (threadIdx.x == 0 && blockIdx.x == 0) {
    float sw = 0.f, sg = 0.f;
    for (int b = 0; b < bs; b++) { sw += wds[b]; sg += gws[b]; }
    out0[0] = 0.1f * (sg / bs) + 0.1f * (-sw / bs);
  }
}

// ============================================================================
extern "C" void kernel_launch(void* const* d_in, const int* in_sizes, int n_in,
                              void* d_out, int out_size, void* d_ws, size_t ws_size,
                              hipStream_t stream) {
  (void)in_sizes; (void)n_in; (void)out_size; (void)ws_size;
  const float* img  = (const float*)d_in[0];   // (32,196,1024)
  const float* tok  = (const float*)d_in[1];   // (32,256,1024)
  const float* tmsk = (const float*)d_in[2];   // (32,256)
  const int*   tdm  = (const int*)d_in[3];     // (32,256,256)
  const int*   irm  = (const int*)d_in[4];     // (32,196,196)
  float* out = (float*)d_out;

  const size_t NM = (size_t)NI * NT;           // 196*256
  const size_t Ncos = (size_t)BSZ * NM;
  const size_t Ncs  = (size_t)BSZ * NI * NI;
  const size_t Nct  = (size_t)BSZ * NT * NT;

  // workspace layout (floats, 64-float aligned)
  float* w = (float*)d_ws;
  size_t o = 0;
  auto alloc = [&](size_t n) { float* p = w + o; o += (n + 63) & ~(size_t)63; return p; };
  float* Xn    = alloc((size_t)BSZ * NI * DD);
  float* Yn    = alloc((size_t)BSZ * NT * DD);
  float* Ccos  = alloc(Ncos);
  float* Cs    = alloc(Ncs);
  float* Ct    = alloc(Nct);
  float* Abuf  = alloc(Ncos);
  float* Qbuf  = alloc(Ncos);
  float* delta = alloc((size_t)BSZ * NI);
  float* sigma = alloc((size_t)BSZ * NT);
  float* M1    = alloc(Ncos);
  float* Cg    = alloc(Ncos);
  float* gamma = alloc(Ncos);
  float* arow  = alloc((size_t)BSZ * NI);
  float* ccol  = alloc((size_t)BSZ * NT);
  float* pmin  = alloc(512);
  float* pmax  = alloc(512);
  float* thr   = alloc(64);
  float* wds   = alloc(64);
  float* gws   = alloc(64);

  float* Twd_out  = out + 1;
  float* Tgwd_out = out + 1 + Ncos;

  // 1) normalize features (tokens masked first)
  k_norm_rows<<<BSZ * NI, TPB, 0, stream>>>(img, nullptr, Xn, DD);
  k_norm_rows<<<BSZ * NT, TPB, 0, stream>>>(tok, tmsk, Yn, DD);

  // 2) cosine-distance matrices via fp32 WMMA:  D = 1 - A.B^T
  dim3 gNM(13, 2, BSZ);   // 196 -> 13 M-tiles, 256 -> 16 N-tiles / 8 waves
  dim3 gNN(13, 2, BSZ);   // 196x196: 13 N-tiles (3 waves idle-masked)
  dim3 gMM(16, 2, BSZ);   // 256x256
  k_wmma_gemm_nt<<<gNM, TPB, 0, stream>>>(Xn, Yn, Ccos, NI, NT, DD, nullptr, nullptr, -1.f, 1.f);
  k_wmma_gemm_nt<<<gNN, TPB, 0, stream>>>(Xn, Xn, Cs,   NI, NI, DD, nullptr, nullptr, -1.f, 1.f);
  k_wmma_gemm_nt<<<gMM, TPB, 0, stream>>>(Yn, Yn, Ct,   NT, NT, DD, nullptr, nullptr, -1.f, 1.f);

  // 3) global threshold-relu (per matrix family), masks applied after relu
  k_minmax_partial<<<512, TPB, 0, stream>>>(Ccos, Ncos, pmin, pmax);
  k_minmax_final<<<1, TPB, 0, stream>>>(pmin, pmax, 512, thr);
  k_thr_relu_mask<<<2048, TPB, 0, stream>>>(Ccos, nullptr, thr, Ncos);

  k_minmax_partial<<<512, TPB, 0, stream>>>(Cs, Ncs, pmin, pmax);
  k_minmax_final<<<1, TPB, 0, stream>>>(pmin, pmax, 512, thr);
  k_thr_relu_mask<<<2048, TPB, 0, stream>>>(Cs, irm, thr, Ncs);

  k_minmax_partial<<<512, TPB, 0, stream>>>(Ct, Nct, pmin, pmax);
  k_minmax_final<<<1, TPB, 0, stream>>>(pmin, pmax, 512, thr);
  k_thr_relu_mask<<<2048, TPB, 0, stream>>>(Ct, tdm, thr, Nct);

  // 4) Wasserstein IPOT on Ccos (20 iterations)
  k_exp_neg2<<<2048, TPB, 0, stream>>>(Ccos, Abuf, Ncos);
  k_fill<<<64, TPB, 0, stream>>>(sigma, 1.0f / NT, (size_t)BSZ * NT);
  for (int it = 0; it < 20; it++) {
    k_ipot_qdelta<<<BSZ * NI, TPB, 0, stream>>>(Abuf, Qbuf, delta, sigma, NI, NT, it == 0);
    k_ipot_sigma<<<BSZ, TPB, 0, stream>>>(Qbuf, delta, sigma, NI, NT);
  }
  k_ipot_finalize<<<BSZ * NI, TPB, 0, stream>>>(Qbuf, delta, sigma, Twd_out, NI, NT);
  k_batch_dot<<<BSZ, TPB, 0, stream>>>(Ccos, Twd_out, wds, NM);

  // 5) Gromov-Wasserstein
  k_rowsumsq<<<BSZ * NI, TPB, 0, stream>>>(Cs, arow, NI, 1.0f / NI);
  k_rowsumsq<<<BSZ * NT, TPB, 0, stream>>>(Ct, ccol, NT, 1.0f / NT);
  k_fill<<<2048, TPB, 0, stream>>>(gamma, 1.0f / (float)(NI * NT), Ncos);

  for (int outer = 0; outer < 5; outer++) {
    // C_gamma = Cst - 2 * Cs . gamma . Ct^T
    k_wmma_gemm_nn<<<gNM, TPB, 0, stream>>>(Cs, gamma, M1, NI, NI, NT);
    k_wmma_gemm_nt<<<gNM, TPB, 0, stream>>>(M1, Ct, Cg, NI, NT, NT, arow, ccol, -2.f, 0.f);
    // gamma = IPOT(C_gamma, 20)
    k_exp_neg2<<<2048, TPB, 0, stream>>>(Cg, Abuf, Ncos);
    k_fill<<<64, TPB, 0, stream>>>(sigma, 1.0f / NT, (size_t)BSZ * NT);
    for (int it = 0; it < 20; it++) {
      k_ipot_qdelta<<<BSZ * NI, TPB, 0, stream>>>(Abuf, Qbuf, delta, sigma, NI, NT, it == 0);
      k_ipot_sigma<<<BSZ, TPB, 0, stream>>>(Qbuf, delta, sigma, NI, NT);
    }
    k_ipot_finalize<<<BSZ * NI, TPB, 0, stream>>>(Qbuf, delta, sigma, gamma, NI, NT);
  }
  // final C_gamma with converged gamma, distance, and T_gwd output
  k_wmma_gemm_nn<<<gNM, TPB, 0, stream>>>(Cs, gamma, M1, NI, NI, NT);
  k_wmma_gemm_nt<<<gNM, TPB, 0, stream>>>(M1, Ct, Cg, NI, NT, NT, arow, ccol, -2.f, 0.f);
  k_batch_dot<<<BSZ, TPB, 0, stream>>>(Cg, gamma, gws, NM);
  (void)hipMemcpyAsync(Tgwd_out, gamma, Ncos * sizeof(float), hipMemcpyDeviceToDevice, stream);

  // 6) scalar twd
  k_final<<<1, 64, 0, stream>>>(wds, gws, out, BSZ);
}